// HausdorffDTLoss_70858370449939
// MI455X (gfx1250) — compile-verified
//
#include <hip/hip_runtime.h>
#include <hip/hip_bf16.h>

#define BB 4
#define CC 4
#define HH 256
#define WW 256
#define HW (HH * WW)          /* 65536 */
#define NIMG (BB * CC)        /* 16 (b,c) images */
#define NARR 4                /* site arrays: pred~fg, predfg, tgt~fg, tgtfg */
#define NLINES (NARR * NIMG * HH) /* 16384 lines per pass */
#define ARRSTRIDE ((size_t)NIMG * HW)
#define TOTE (CC * BB * HW)   /* 1048576 elements in final sum */
#define BIGF 1.0e9f
#define NPART 1024

typedef __attribute__((ext_vector_type(2))) float v2f;
typedef __attribute__((ext_vector_type(8))) float v8f;

// ---------------------------------------------------------------------------
// K1: softmax over channels + build the 4 site arrays per (b,c) image.
// f = 0 where "site", BIG elsewhere (exactly as the reference's jnp.where).
__global__ void k_sites(const float* __restrict__ pred, const int* __restrict__ tgt,
                        float* __restrict__ F) {
  int gid = blockIdx.x * blockDim.x + threadIdx.x;   // 0 .. B*HW-1
  int b   = gid >> 16;           // / HW
  int pix = gid & (HW - 1);
  const float* pb = pred + (size_t)b * CC * HW + pix;
  float x0 = pb[0], x1 = pb[HW], x2 = pb[2 * HW], x3 = pb[3 * HW];
  float mx = fmaxf(fmaxf(x0, x1), fmaxf(x2, x3));
  float e0 = expf(x0 - mx), e1 = expf(x1 - mx), e2 = expf(x2 - mx), e3 = expf(x3 - mx);
  float inv = 1.0f / (e0 + e1 + e2 + e3);
  float p[CC] = {e0 * inv, e1 * inv, e2 * inv, e3 * inv};
  int t = tgt[gid];
#pragma unroll
  for (int c = 0; c < CC; ++c) {
    size_t o = (size_t)(b * CC + c) * HW + pix;
    bool fg = p[c] > 0.5f;
    bool tc = (t == c);
    F[0 * ARRSTRIDE + o] = fg ? BIGF : 0.0f;  // sites = ~fg  -> fg_dist^2
    F[1 * ARRSTRIDE + o] = fg ? 0.0f : BIGF;  // sites =  fg  -> bg_dist^2
    F[2 * ARRSTRIDE + o] = tc ? BIGF : 0.0f;  // sites = ~t
    F[3 * ARRSTRIDE + o] = tc ? 0.0f : BIGF;  // sites =  t
  }
}

// ---------------------------------------------------------------------------
// K2: 1D min-plus DT along rows.  One block per line; line staged in LDS,
// every lane broadcast-reads lds[j] (bank-conflict-free).
__global__ void k_dt_rows(const float* __restrict__ F, float* __restrict__ G) {
  __shared__ float lds[WW];
  size_t base = (size_t)blockIdx.x * WW;   // lines are contiguous in F
  int i = threadIdx.x;
  lds[i] = F[base + i];
  __syncthreads();
  float m = 3.0e38f;
#pragma unroll 8
  for (int j = 0; j < WW; ++j) {
    float dj = (float)(i - j);
    m = fminf(m, lds[j] + dj * dj);
  }
  G[base + i] = m;
}

// ---------------------------------------------------------------------------
// K3: 1D min-plus DT along columns.  Block = (image, out-row); lane = column.
// Loads are coalesced; whole working set is L2-resident (33 MB << 192 MB).
__global__ void k_dt_cols(const float* __restrict__ G, float* __restrict__ F) {
  int img = blockIdx.x >> 8;        // / HH
  int r   = blockIdx.x & (HH - 1);
  int c   = threadIdx.x;
  const float* g = G + (size_t)img * HW + c;
  float m = 3.0e38f;
#pragma unroll 8
  for (int rp = 0; rp < HH; ++rp) {
    float dr = (float)(r - rp);
    m = fminf(m, g[(size_t)rp * WW] + dr * dr);
  }
  F[(size_t)img * HW + (size_t)r * WW + c] = m;
}

// ---------------------------------------------------------------------------
// K4: per-element loss term + deterministic reduction.
// Wave-level sum of the 32 per-lane accumulators is done with
// v_wmma_f32_16x16x4_f32 against an all-ones B matrix: D rows hold row-sums
// of A (replicated across N); summing the 8 D VGPRs + one shfl_xor(16)
// yields the full wave sum in every lane.
__global__ void k_loss_partial(const float* __restrict__ pred, const int* __restrict__ tgt,
                               const float* __restrict__ F, float* __restrict__ partial) {
  int tid = threadIdx.x;
  int gid = blockIdx.x * blockDim.x + tid;
  int stride = gridDim.x * blockDim.x;     // 1024*256 -> exactly 4 iters/thread
  float acc = 0.0f;
  for (int e = gid; e < TOTE; e += stride) {
    int img = e >> 16;                 // / HW
    int pix = e & (HW - 1);
    int b = img >> 2, c = img & 3;
    const float* pb = pred + (size_t)b * CC * HW + pix;
    float x0 = pb[0], x1 = pb[HW], x2 = pb[2 * HW], x3 = pb[3 * HW];
    float mx = fmaxf(fmaxf(x0, x1), fmaxf(x2, x3));
    float e0 = expf(x0 - mx), e1 = expf(x1 - mx), e2 = expf(x2 - mx), e3 = expf(x3 - mx);
    float xc = (c == 0) ? e0 : (c == 1) ? e1 : (c == 2) ? e2 : e3;
    float pc = xc / (e0 + e1 + e2 + e3);
    float tv = (tgt[(size_t)b * HW + pix] == c) ? 1.0f : 0.0f;
    size_t o = (size_t)img * HW + pix;
    float a0 = F[0 * ARRSTRIDE + o];   // pred: dist^2 to ~fg sites
    float a1 = F[1 * ARRSTRIDE + o];   // pred: dist^2 to  fg sites
    float a2 = F[2 * ARRSTRIDE + o];   // tgt:  dist^2 to ~t sites
    float a3 = F[3 * ARRSTRIDE + o];   // tgt:  dist^2 to  t sites
    // any_fg inference: with >=1 fg site every pixel's a_bg <= 130050,
    // with none it is exactly 1e9.
    float pdt = (a1 < 1.0e8f) ? (sqrtf(a0) + sqrtf(a1)) : 0.0f;
    float tdt = (a3 < 1.0e8f) ? (sqrtf(a2) + sqrtf(a3)) : 0.0f;
    float pe = pc - tv;
    acc += (pe * pe) * (pdt * pdt + tdt * tdt);
  }

  // ---- wave32 reduction via WMMA f32 16x16x4 (EXEC is all-ones here) ----
  v2f a;  a[0] = acc;  a[1] = 0.0f;       // each lane contributes one A element
  v2f bo; bo[0] = 1.0f; bo[1] = 1.0f;     // B = all ones (4x16)
  v8f cz = {0.f, 0.f, 0.f, 0.f, 0.f, 0.f, 0.f, 0.f};
  v8f d = __builtin_amdgcn_wmma_f32_16x16x4_f32(
      /*neg_a=*/false, a, /*neg_b=*/false, bo,
      /*c_mod=*/(short)0, cz, /*reuse_a=*/false, /*reuse_b=*/false);
  float s8 = d[0] + d[1] + d[2] + d[3] + d[4] + d[5] + d[6] + d[7];
  float tot = s8 + __shfl_xor(s8, 16, 32);   // combine the two lane halves

  __shared__ float wsum[8];
  int lane = tid & 31, wid = tid >> 5;
  if (lane == 0) wsum[wid] = tot;
  __syncthreads();
  if (tid == 0) {
    float bsum = 0.0f;
#pragma unroll
    for (int w = 0; w < 8; ++w) bsum += wsum[w];
    partial[blockIdx.x] = bsum;
  }
}

// ---------------------------------------------------------------------------
// K5: deterministic final sum of block partials.
__global__ void k_finish(const float* __restrict__ partial, float* __restrict__ out) {
  __shared__ float lds[256];
  int tid = threadIdx.x;
  float s = 0.0f;
  for (int i = tid; i < NPART; i += 256) s += partial[i];
  lds[tid] = s;
  __syncthreads();
  if (tid == 0) {
    float tot = 0.0f;
    for (int i = 0; i < 256; ++i) tot += lds[i];
    out[0] = tot / (float)TOTE;   // (sum over classes of means) / NCLASSES
  }
}

// ---------------------------------------------------------------------------
extern "C" void kernel_launch(void* const* d_in, const int* in_sizes, int n_in,
                              void* d_out, int out_size, void* d_ws, size_t ws_size,
                              hipStream_t stream) {
  (void)in_sizes; (void)n_in; (void)out_size; (void)ws_size;
  const float* pred = (const float*)d_in[0];   // (4,4,256,256) f32
  const int*   tgt  = (const int*)d_in[1];     // (4,256,256) i32
  float* out = (float*)d_out;

  float* F       = (float*)d_ws;                          // 4*16*HW floats (16 MB)
  float* G       = F + NARR * ARRSTRIDE;                  // 16 MB ping-pong
  float* partial = G + NARR * ARRSTRIDE;                  // NPART floats

  k_sites       <<<(BB * HW) / 256, 256, 0, stream>>>(pred, tgt, F);
  k_dt_rows     <<<NLINES,          256, 0, stream>>>(F, G);
  k_dt_cols     <<<NLINES,          256, 0, stream>>>(G, F);
  k_loss_partial<<<NPART,           256, 0, stream>>>(pred, tgt, F, partial);
  k_finish      <<<1,               256, 0, stream>>>(partial, out);
}